// SlimeNetKeras_2224793059471
// MI455X (gfx1250) — compile-verified
//
#include <hip/hip_runtime.h>

// Problem constants (match reference)
#define C_CH   16
#define H_DIM  512
#define W_DIM  512
#define HW     (H_DIM * W_DIM)
#define HID    128
#define KF     80            // 5*C gathered features
#define NSTEPS 8

// Tiling
#define TP     128           // pixels per block (one row segment)
#define NTHR   256           // 8 wave32
#define NWAVE  8

// LDS layout (float offsets). All v2f-accessed offsets are even -> 8B aligned.
#define XS      82                       // padded X row stride (floats)
#define OFF_X   0
#define SZ_X    (TP * XS)                // 10496
#define OFF_W1  (OFF_X + SZ_X)           // 10496
#define SZ_W1   (20 * 8 * 32 * 2)        // 10240 floats (float2 fragments)
#define OFF_W2  (OFF_W1 + SZ_W1)         // 20736
#define SZ_W2   (32 * 32 * 2)            // 2048
#define OFF_H   (OFF_W2 + SZ_W2)         // 22784
#define HSTRIDE 17                       // Hs[k][m], k-stride 17 -> conflict-free reads
#define SZ_HW_  (HID * HSTRIDE)          // 2176 per wave
#define SZ_H    (NWAVE * SZ_HW_)         // 17408
#define OFF_D   (OFF_H + SZ_H)           // 40192
#define DSTRIDE 130
#define SZ_D    (C_CH * DSTRIDE)         // 2080
#define SMEM_FLOATS (OFF_D + SZ_D)       // 42272
#define SMEM_BYTES  (SMEM_FLOATS * 4)    // 169088 B < 320 KB

typedef float v2f __attribute__((ext_vector_type(2)));
typedef float v8f __attribute__((ext_vector_type(8)));

__global__ __launch_bounds__(NTHR, 1)
void nca_step_kernel(const float* __restrict__ src, float* __restrict__ dst,
                     const float* __restrict__ W1, const float* __restrict__ b1,
                     const float* __restrict__ W2, const float* __restrict__ b2) {
    extern __shared__ float smem[];

    const int tid  = threadIdx.x;
    const int lane = tid & 31;
    const int wave = tid >> 5;
    const int half = lane >> 4;   // 0/1: which K-pair this lane holds
    const int l16  = lane & 15;

    const int y  = blockIdx.y;
    const int x0 = blockIdx.x * TP;
    const int yu = (y == 0) ? 0 : y - 1;            // clamp low
    const int yd = (y == H_DIM - 1) ? 0 : y + 1;    // wrap high (reference quirk)

    // ---------------- Stage 1: cooperative LDS staging ----------------
    // Gathered feature tile X[m][f], f = neighbor_block*16 + channel
    #pragma unroll
    for (int i = 0; i < (KF * TP) / NTHR; ++i) {          // 40 iters
        int idx = tid + NTHR * i;                          // 0..10239
        int f = idx >> 7;                                  // 0..79
        int m = idx & 127;
        int b = f >> 4;                                    // 0:center 1:up 2:down 3:left 4:right
        int c = f & 15;
        int x = x0 + m;
        int yy = y, xx = x;
        if (b == 1)      yy = yu;
        else if (b == 2) yy = yd;
        else if (b == 3) xx = (x == 0) ? 0 : x - 1;                 // clamp low
        else if (b == 4) xx = (x == W_DIM - 1) ? 0 : x + 1;         // wrap high
        smem[OFF_X + m * XS + f] = src[c * HW + yy * W_DIM + xx];
    }

    // W1 pre-swizzled into B-fragment layout: frag[(kk*8+j)*32 + lane] =
    //   { W1[4kk+2*half][16j+l16], W1[4kk+2*half+1][16j+l16] }
    {
        v2f* w1f = (v2f*)&smem[OFF_W1];
        #pragma unroll
        for (int i = 0; i < (20 * 8 * 32) / NTHR; ++i) {   // 20 iters
            int idx = tid + NTHR * i;                      // 0..5119
            int kk = idx >> 8;
            int rem = idx & 255;
            int j = rem >> 5;
            int l = rem & 31;
            int k0 = kk * 4 + 2 * (l >> 4);
            int n = j * 16 + (l & 15);
            v2f w;
            w.x = W1[k0 * HID + n];
            w.y = W1[(k0 + 1) * HID + n];
            w1f[idx] = w;
        }
        v2f* w2f = (v2f*)&smem[OFF_W2];
        #pragma unroll
        for (int i = 0; i < (32 * 32) / NTHR; ++i) {       // 4 iters
            int idx = tid + NTHR * i;                      // 0..1023
            int kk = idx >> 5;
            int l = idx & 31;
            int k0 = kk * 4 + 2 * (l >> 4);
            v2f w;
            w.x = W2[k0 * C_CH + (l & 15)];
            w.y = W2[(k0 + 1) * C_CH + (l & 15)];
            w2f[idx] = w;
        }
    }

    __syncthreads();

    // ---------------- Stage 2: per-wave 16-pixel M-tile ----------------
    const int m0 = wave * 16;           // pixel offset within block tile
    float* hs = &smem[OFF_H + wave * SZ_HW_];

    // Layer 1: X(16x80) @ W1(80x128) + b1  -> 8 N-tiles of 16x16, K in steps of 4
    v8f acc[8];
    #pragma unroll
    for (int j = 0; j < 8; ++j) {
        float bv = b1[j * 16 + l16];
        #pragma unroll
        for (int r = 0; r < 8; ++r) acc[j][r] = bv;
    }
    const v2f* w1f = (const v2f*)&smem[OFF_W1];
    #pragma unroll 4
    for (int kk = 0; kk < 20; ++kk) {
        // A-fragment: lane holds X[m0+l16][4kk + 2*half (+1)]
        v2f a = *(const v2f*)&smem[OFF_X + (m0 + l16) * XS + kk * 4 + 2 * half];
        #pragma unroll
        for (int j = 0; j < 8; ++j) {
            v2f b = w1f[(kk * 8 + j) * 32 + lane];
            acc[j] = __builtin_amdgcn_wmma_f32_16x16x4_f32(
                false, a, false, b, (short)0, acc[j], false, false);
        }
    }

    // ReLU + stash hidden transposed: Hs[n][m], n = hidden idx (layer-2 K)
    #pragma unroll
    for (int j = 0; j < 8; ++j) {
        #pragma unroll
        for (int r = 0; r < 8; ++r) {
            float v = fmaxf(acc[j][r], 0.0f);
            hs[(j * 16 + l16) * HSTRIDE + r + 8 * half] = v;
        }
    }

    // Layer 2: H(16x128) @ W2(128x16) + b2
    v8f acc2;
    {
        float bv = b2[l16];
        #pragma unroll
        for (int r = 0; r < 8; ++r) acc2[r] = bv;
    }
    const v2f* w2f = (const v2f*)&smem[OFF_W2];
    #pragma unroll 8
    for (int kk = 0; kk < 32; ++kk) {
        int k0 = kk * 4 + 2 * half;
        v2f a;
        a.x = hs[k0 * HSTRIDE + l16];
        a.y = hs[(k0 + 1) * HSTRIDE + l16];
        v2f b = w2f[kk * 32 + lane];
        acc2 = __builtin_amdgcn_wmma_f32_16x16x4_f32(
            false, a, false, b, (short)0, acc2, false, false);
    }

    // Delta -> LDS transpose buffer Ds[c][pixel]
    #pragma unroll
    for (int r = 0; r < 8; ++r) {
        smem[OFF_D + l16 * DSTRIDE + m0 + r + 8 * half] = acc2[r];
    }

    __syncthreads();

    // ---------------- Stage 3: coalesced residual add + store ----------------
    #pragma unroll
    for (int i = 0; i < (C_CH * TP) / NTHR; ++i) {         // 8 iters
        int idx = tid + NTHR * i;                          // 0..2047
        int c = idx >> 7;
        int m = idx & 127;
        int g = c * HW + y * W_DIM + (x0 + m);
        float sold = src[g];
        float v = (c == 0) ? sold : (sold + smem[OFF_D + c * DSTRIDE + m]);
        dst[g] = v;
    }
}

extern "C" void kernel_launch(void* const* d_in, const int* in_sizes, int n_in,
                              void* d_out, int out_size, void* d_ws, size_t ws_size,
                              hipStream_t stream) {
    const float* state = (const float*)d_in[0];
    const float* W1    = (const float*)d_in[1];
    const float* b1    = (const float*)d_in[2];
    const float* W2    = (const float*)d_in[3];
    const float* b2    = (const float*)d_in[4];
    // d_in[5] = n_steps (device scalar); reference fixes N_STEPS = 8.

    float* out = (float*)d_out;
    float* ws  = (float*)d_ws;   // needs C*H*W*4 = 16 MB

    dim3 grid(W_DIM / TP, H_DIM);
    const float* src = state;
    for (int i = 0; i < NSTEPS; ++i) {
        float* dst = (i & 1) ? out : ws;   // step 7 (last) lands in d_out
        nca_step_kernel<<<grid, NTHR, SMEM_BYTES, stream>>>(src, dst, W1, b1, W2, b2);
        src = dst;
    }
}